// ProtoSimModel_48412871360945
// MI455X (gfx1250) — compile-verified
//
#include <hip/hip_runtime.h>
#include <hip/hip_bf16.h>
#include <math.h>

// ---------------------------------------------------------------------------
// ProtoSimModel for MI455X (gfx1250, wave32).
//   sim[b,p] = (x_b . proto_p) / max(||x_b||,1e-12)   (protos pre-normalized)
//   argmax(dist) == argmin(sim)  (dist = 1/(1+exp(sim)/10) is decreasing)
//   predict_role is a 20x13 table lookup (only 20 distinct "nearest" rows).
// GEMM done with V_WMMA_F32_16X16X4_F32 (fp32-exact, matches reference).
// Bandwidth-bound: single streaming pass over x (402 MB -> ~17us @ 23.3TB/s).
// ---------------------------------------------------------------------------

typedef __attribute__((ext_vector_type(2))) float v2f;
typedef __attribute__((ext_vector_type(8))) float v8f;

#define RROLE 13
#define DDIM  768
#define PPROT 20
#define KSTEPS (DDIM / 4)   // 192 WMMA k-steps of 4

// ws layout: [ Bs : 2 tiles * 192 ksteps * 32 lanes * 2 floats = 24576 f ]
//            [ logits : 20*13 floats ]
#define BS_FLOATS (2 * KSTEPS * 64)

// ---------------------------------------------------------------------------
// Kernel 1: prototype prep (1 block, 256 threads). Negligible cost.
//  - Bs swizzled for WMMA B fragments: for tile t, kstep g, lane l:
//      n = t*16 + (l&15), K = g*4 + (l>>4)*2 + j  (j=0,1)
//      Bs[((t*KSTEPS + g)*32 + l)*2 + j] = protoN[n][K]   (0 for n>=20)
//    This mirrors the ISA 32-bit 4x16 B layout (VGPR0: K={0,2}, VGPR1: K={1,3},
//    lanes 0-15 / 16-31 split), giving each lane one contiguous 8B load/step.
// ---------------------------------------------------------------------------
__global__ void proto_prep_kernel(const float* __restrict__ prot,
                                  const int* __restrict__ role_id,
                                  const float* __restrict__ W,
                                  const float* __restrict__ bias,
                                  float* __restrict__ Bs,
                                  float* __restrict__ logits) {
  __shared__ float inv_norm[32];
  const int tid = threadIdx.x;
  const int rid = *role_id;
  const float* pr = prot + (size_t)rid * PPROT * DDIM;

  if (tid < 32) {
    float s = 0.f;
    if (tid < PPROT) {
      const float* row = pr + tid * DDIM;
      for (int d = 0; d < DDIM; ++d) { float v = row[d]; s = fmaf(v, v, s); }
    }
    inv_norm[tid] = (tid < PPROT) ? 1.0f / fmaxf(sqrtf(s), 1e-12f) : 0.0f;
  }
  __syncthreads();

  // swizzled, zero-padded B operand
  for (int idx = tid; idx < 2 * KSTEPS * 32; idx += blockDim.x) {
    const int lane = idx & 31;
    const int g    = (idx >> 5) % KSTEPS;
    const int t    = idx / (KSTEPS * 32);
    const int n    = t * 16 + (lane & 15);
    const int k    = g * 4 + (lane >> 4) * 2;
    float v0 = 0.f, v1 = 0.f;
    if (n < PPROT) {
      const float inv = inv_norm[n];
      const float* row = pr + n * DDIM;
      v0 = row[k] * inv;
      v1 = row[k + 1] * inv;
    }
    Bs[idx * 2 + 0] = v0;
    Bs[idx * 2 + 1] = v1;
  }

  // logits[p][r] = bias[r] + protoN[p] . W[r]
  for (int idx = tid; idx < PPROT * RROLE; idx += blockDim.x) {
    const int p = idx / RROLE, r = idx % RROLE;
    const float* row = pr + p * DDIM;
    const float* w   = W + r * DDIM;
    float acc = 0.f;
    for (int d = 0; d < DDIM; ++d) acc = fmaf(row[d], w[d], acc);
    logits[idx] = acc * inv_norm[p] + bias[r];
  }
}

// ---------------------------------------------------------------------------
// Kernel 2: one wave32 per 16-row tile. 8 waves / 256-thread block.
// A fragment (ISA 16x4 f32 layout): lane l supplies A[M=l&15][K=k0+(l>>4)*2+j].
// ---------------------------------------------------------------------------
__global__ void proto_sim_kernel(const float* __restrict__ x,
                                 const float* __restrict__ Bs,
                                 const float* __restrict__ logits,
                                 float* __restrict__ out,
                                 int B, int ntiles) {
  const int lane = threadIdx.x & 31;
  const int wave = threadIdx.x >> 5;
  const int tile = blockIdx.x * (blockDim.x >> 5) + wave;
  if (tile >= ntiles) return;

  const int base = tile << 4;
  const int half = lane >> 4;     // 0: lanes 0-15, 1: lanes 16-31
  const int col  = lane & 15;

  const float* px  = x + (size_t)(base + col) * DDIM + half * 2;
  const float* b0p = Bs + lane * 2;                    // protos 0-15
  const float* b1p = Bs + KSTEPS * 64 + lane * 2;      // protos 16-31 (padded)

  v8f c0 = {};
  v8f c1 = {};
  float sq = 0.f;   // per-lane partial sum-of-squares of this lane's K slice

#pragma unroll 4
  for (int g = 0; g < KSTEPS; ++g) {
    v2f a  = *(const v2f*)(px + g * 4);
    v2f b0 = *(const v2f*)(b0p + g * 64);
    v2f b1 = *(const v2f*)(b1p + g * 64);
    sq = fmaf(a.x, a.x, sq);
    sq = fmaf(a.y, a.y, sq);
    c0 = __builtin_amdgcn_wmma_f32_16x16x4_f32(false, a, false, b0,
                                               (short)0, c0, false, false);
    c1 = __builtin_amdgcn_wmma_f32_16x16x4_f32(false, a, false, b1,
                                               (short)0, c1, false, false);
  }

  // row sum-of-squares: lanes l and l^16 cover K%4 ∈ {0,1} and {2,3} of row l&15
  const float rsq = sq + __shfl_xor(sq, 16, 32);   // lane l holds S[l&15]

#pragma unroll
  for (int i = 0; i < 8; ++i) {
    // C layout: VGPR i = row (i + 8*half), column = col
    float v0 = c0[i];
    float v1 = (col < (PPROT - 16)) ? c1[i] : 3.0e38f;  // mask padded protos
    float v;
    int   idx;
    if (v1 < v0) { v = v1; idx = 16 + col; }
    else         { v = v0; idx = col; }

    // butterfly argmin over the 16 lanes of each half; tie -> lower index
    // (matches jnp.argmax first-occurrence on the monotone-mapped dist)
#pragma unroll
    for (int s = 1; s < 16; s <<= 1) {
      const float ov = __shfl_xor(v, s, 32);
      const int   oi = __shfl_xor(idx, s, 32);
      if (ov < v || (ov == v && oi < idx)) { v = ov; idx = oi; }
    }

    // fetch this row's norm (uniform shuffles: no divergence here)
    const float S_lo = __shfl(rsq, i, 32);
    const float S_hi = __shfl(rsq, i + 8, 32);
    const float S    = half ? S_hi : S_lo;

    const float sim  = v / fmaxf(sqrtf(S), 1e-12f);
    const float e    = expf(sim) * 0.1f;
    const float dist = 1.0f / (1.0f + e);

    if (col == 0) {
      const int row = base + i + half * 8;
      if (row < B) {
        out[row] = dist;                           // min_dist
        float* pd = out + B + (size_t)row * RROLE; // predict_role row
        const float* lg = logits + idx * RROLE;
#pragma unroll
        for (int r = 0; r < RROLE; ++r) pd[r] = lg[r];
      }
    }
  }
}

// ---------------------------------------------------------------------------
extern "C" void kernel_launch(void* const* d_in, const int* in_sizes, int n_in,
                              void* d_out, int out_size, void* d_ws, size_t ws_size,
                              hipStream_t stream) {
  const float* x      = (const float*)d_in[0];   // [B, 768]
  const int*   roleid = (const int*)d_in[1];     // scalar
  const float* prot   = (const float*)d_in[2];   // [260, 768]
  const float* W      = (const float*)d_in[3];   // [13, 768]
  const float* bias   = (const float*)d_in[4];   // [13]
  float* out = (float*)d_out;                    // [B] ++ [B,13]

  const int B = in_sizes[0] / DDIM;

  float* Bs     = (float*)d_ws;
  float* logits = Bs + BS_FLOATS;

  proto_prep_kernel<<<1, 256, 0, stream>>>(prot, roleid, W, bias, Bs, logits);

  const int ntiles = (B + 15) / 16;
  const int blocks = (ntiles + 7) / 8;           // 8 waves per block
  proto_sim_kernel<<<blocks, 256, 0, stream>>>(x, Bs, logits, out, B, ntiles);
}